// Conv2d_69398081568838
// MI455X (gfx1250) — compile-verified
//
#include <hip/hip_runtime.h>

typedef __attribute__((ext_vector_type(2))) float v2f;
typedef __attribute__((ext_vector_type(4))) float v4f;
typedef __attribute__((ext_vector_type(8))) float v8f;
typedef __attribute__((ext_vector_type(4))) unsigned int u32x4;
typedef __attribute__((ext_vector_type(8))) int i32x8;
typedef __attribute__((ext_vector_type(4))) int i32x4;

#define INC   128
#define OUTC  256
#define INH   112
#define INW   112
#define OHH   110
#define OWW   110
#define HW_IN  (INH * INW)          // 12544
#define HW_OUT (OHH * OWW)          // 12100
#define GK    (INC * 9)             // 1152, GEMM K
#define KSTEP 32                    // K panel staged in LDS
#define NP    (GK / KSTEP)          // 36 panels
#define MT    128                   // M tile (out channels) per workgroup
#define NT    64                    // N tile (spatial positions) per workgroup
#define LSTRIDE 36                  // padded LDS row stride (dwords): 32 + 4

#if defined(__has_builtin)
#if __has_builtin(__builtin_amdgcn_tensor_load_to_lds) && __has_builtin(__builtin_amdgcn_s_wait_tensorcnt)
#define USE_TDM 1
#endif
#endif
#ifndef USE_TDM
#define USE_TDM 0
#endif

#if USE_TDM
// TDM: DMA one 128x32 f32 weight panel (row stride GK dwords) into LDS with a
// padded destination row stride of 36 dwords (pad every 32 dwords by 4 dwords),
// so column-direction fragment reads hit distinct banks.
__device__ static inline void tdm_load_A(const float* gptr, unsigned lds_off) {
    const unsigned TD0 = 1u << 20, TD1 = 1u << 20;  // huge tensor dims: no OOB clipping
    unsigned long long ga = (unsigned long long)(uintptr_t)gptr;
    u32x4 g0;
    g0[0] = 1u;                                     // count=1 (valid user descriptor)
    g0[1] = lds_off;                                // lds_addr (bytes)
    g0[2] = (unsigned)(ga & 0xffffffffu);           // global_addr[31:0]
    g0[3] = (unsigned)((ga >> 32) & 0x1ffffffu)     // global_addr[56:32]
          | (2u << 30);                             // type = 2 ("image")
    i32x8 g1;
    g1[0] = (int)((2u << 16)                        // data_size = 4 bytes
                | (1u << 20)                        // pad_enable
                | (4u << 22)                        // pad_interval: 32 dwords
                | (3u << 25));                      // pad_amount: 4 dwords
    g1[1] = (int)((TD0 & 0xffffu) << 16);           // tensor_dim0[15:0]
    g1[2] = (int)(((TD0 >> 16) & 0xffffu) | ((TD1 & 0xffffu) << 16));
    g1[3] = (int)(((TD1 >> 16) & 0xffffu) | (32u << 16)); // tile_dim0 = 32
    g1[4] = MT;                                     // tile_dim1 = 128, tile_dim2 = 0
    g1[5] = GK;                                     // tensor_dim0_stride = 1152
    g1[6] = 0;
    g1[7] = 0;
    i32x4 z4 = {0, 0, 0, 0};
#if defined(__clang_major__) && (__clang_major__ >= 23)
    i32x8 z8 = {0, 0, 0, 0, 0, 0, 0, 0};
    __builtin_amdgcn_tensor_load_to_lds(g0, g1, z4, z4, z8, 0);
#else
    __builtin_amdgcn_tensor_load_to_lds(g0, g1, z4, z4, 0);
#endif
}
#endif

// Implicit-GEMM conv with V_WMMA_F32_16X16X4_F32 (exact fp32).
// WG = 256 threads = 8 wave32s in 4(M) x 2(N); each wave owns a 32x32 C slab
// (four 16x16 f32 accumulators, 2x2 operand reuse -> 4 WMMAs per 2 A + 2 B loads).
// Double-buffered LDS; TDM stages the weight panel; software-pipelined im2col
// gather stages the transposed input panel; one barrier per K panel.
__global__ __launch_bounds__(256) void conv3x3_wmma_f32(
    const float* __restrict__ x,      // (32,128,112,112)
    const float* __restrict__ w,      // (256,1152) along K
    const float* __restrict__ bias,   // (256,)
    float* __restrict__ out)          // (32,256,110,110)
{
    __shared__ float Alds[2][MT][LSTRIDE];   // 2 x 128 x 36 dwords (TDM-padded rows)
    __shared__ float Bt[2][NT][LSTRIDE];     // 2 x 64 x 36 dwords, [n][k] transposed

    const int tid    = threadIdx.x;
    const int lane   = tid & 31;
    const int wv     = tid >> 5;
    const int lane15 = lane & 15;
    const int khalf  = lane >> 4;
    const int koff   = khalf * 2;

    const int oc0  = blockIdx.y * MT;
    const int pos0 = blockIdx.x * NT;

    const int m_base = (wv & 3) * 32;
    const int n_base = (wv >> 2) * 32;

    // ---- init accumulators with bias ----
    v8f c00, c01, c10, c11;
    #pragma unroll
    for (int j = 0; j < 8; ++j) {
        float b0 = bias[oc0 + m_base + khalf * 8 + j];
        float b1 = bias[oc0 + m_base + 16 + khalf * 8 + j];
        c00[j] = b0; c01[j] = b0;
        c10[j] = b1; c11[j] = b1;
    }

    // ---- B (im2col) gather map: thread t -> one position, 8 consecutive k ----
    const int n_loc   = tid >> 2;        // 0..63
    const int k_chunk = (tid & 3) * 8;   // 0,8,16,24
    int basep;                           // x offset of (n, oh, ow), channel 0, tap (0,0)
    {
        int p  = pos0 + n_loc;
        int n  = p / HW_OUT;
        int r  = p - n * HW_OUT;
        int oh = r / OWW;
        int ow = r - oh * OWW;
        basep = n * (INC * HW_IN) + oh * INW + ow;
    }

    const float* wbase = w + (size_t)oc0 * GK;

#if !USE_TDM
    const int a_row = tid >> 1;          // 0..127
    const int a_col = (tid & 1) * 16;    // 0,16
#endif

    v4f breg[2];
    // gather 8 consecutive-k elements for this thread's position into breg
    auto gather_B = [&](int k0) {
        #pragma unroll
        for (int i = 0; i < 8; ++i) {
            int kg = k0 + k_chunk + i;
            int cc = kg / 9;
            int rs = kg - cc * 9;
            int rr = rs / 3;
            int ss = rs - rr * 3;
            breg[i >> 2][i & 3] = x[basep + cc * HW_IN + rr * INW + ss];
        }
    };

    // ---- prologue: stage panel 0 ----
#if USE_TDM
    if (wv == 0) tdm_load_A(wbase, (unsigned)(uintptr_t)&Alds[0][0][0]);
#endif
    gather_B(0);

    const int arow0 = m_base + lane15;
    const int arow1 = m_base + 16 + lane15;
    const int brow0 = n_base + lane15;
    const int brow1 = n_base + 16 + lane15;

    // ================= pipelined K loop: 36 panels =================
    for (int p = 0; p < NP; ++p) {
        const int buf = p & 1;

        // commit staged B regs (two b128 stores into transposed panel)
        *(v4f*)&Bt[buf][n_loc][k_chunk]     = breg[0];
        *(v4f*)&Bt[buf][n_loc][k_chunk + 4] = breg[1];
#if !USE_TDM
        {
            const float* src = wbase + (size_t)a_row * GK + p * KSTEP + a_col;
            *(v4f*)&Alds[buf][a_row][a_col]      = *(const v4f*)(src);
            *(v4f*)&Alds[buf][a_row][a_col + 4]  = *(const v4f*)(src + 4);
            *(v4f*)&Alds[buf][a_row][a_col + 8]  = *(const v4f*)(src + 8);
            *(v4f*)&Alds[buf][a_row][a_col + 12] = *(const v4f*)(src + 12);
        }
#endif
#if USE_TDM
        __builtin_amdgcn_s_wait_tensorcnt(0);   // A panel p landed (no-op for waves 1..7)
#endif
        __syncthreads();

        // kick off panel p+1: TDM for A, register-staged gather for B
        if (p + 1 < NP) {
            const int k0n = (p + 1) * KSTEP;
#if USE_TDM
            if (wv == 0)
                tdm_load_A(wbase + k0n, (unsigned)(uintptr_t)&Alds[buf ^ 1][0][0]);
#endif
            gather_B(k0n);
        }

        // compute: 8 K-steps of 4; 2 A + 2 B b64 loads feed 4 WMMAs (2x2 reuse)
        #pragma unroll
        for (int kq = 0; kq < KSTEP; kq += 4) {
            v2f a0 = *(const v2f*)&Alds[buf][arow0][kq + koff];
            v2f a1 = *(const v2f*)&Alds[buf][arow1][kq + koff];
            v2f b0 = *(const v2f*)&Bt[buf][brow0][kq + koff];
            v2f b1 = *(const v2f*)&Bt[buf][brow1][kq + koff];
            c00 = __builtin_amdgcn_wmma_f32_16x16x4_f32(
                      false, a0, false, b0, (short)0, c00, false, false);
            c01 = __builtin_amdgcn_wmma_f32_16x16x4_f32(
                      false, a0, false, b1, (short)0, c01, false, false);
            c10 = __builtin_amdgcn_wmma_f32_16x16x4_f32(
                      false, a1, false, b0, (short)0, c10, false, false);
            c11 = __builtin_amdgcn_wmma_f32_16x16x4_f32(
                      false, a1, false, b1, (short)0, c11, false, false);
        }
        // no trailing barrier: next iteration writes the other buffer; its
        // __syncthreads (preceded by s_wait_dscnt) protects buffer reuse.
    }

    // ================= epilogue: scatter C to NCHW output =================
    #pragma unroll
    for (int f = 0; f < 2; ++f) {
        int p  = pos0 + n_base + f * 16 + lane15;
        int n  = p / HW_OUT;
        int r  = p - n * HW_OUT;
        int oh = r / OWW;
        int ow = r - oh * OWW;
        size_t base = (size_t)n * OUTC * HW_OUT + (size_t)oh * OWW + ow;
        v8f cl = f ? c01 : c00;
        v8f ch = f ? c11 : c10;
        #pragma unroll
        for (int j = 0; j < 8; ++j) {
            int m0 = oc0 + m_base + khalf * 8 + j;
            out[base + (size_t)m0 * HW_OUT]        = cl[j];
            out[base + (size_t)(m0 + 16) * HW_OUT] = ch[j];
        }
    }
}

extern "C" void kernel_launch(void* const* d_in, const int* in_sizes, int n_in,
                              void* d_out, int out_size, void* d_ws, size_t ws_size,
                              hipStream_t stream) {
    (void)in_sizes; (void)n_in; (void)out_size; (void)d_ws; (void)ws_size;
    const float* x    = (const float*)d_in[0];
    const float* w    = (const float*)d_in[1];
    const float* bias = (const float*)d_in[2];
    float* out        = (float*)d_out;

    dim3 grid(6050, 2, 1);   // N tiles: 387200/64 ; M tiles: 256/128
    dim3 block(256, 1, 1);
    conv3x3_wmma_f32<<<grid, block, 0, stream>>>(x, w, bias, out);
}